// Top1Gate_6536940225050
// MI455X (gfx1250) — compile-verified
//
#include <hip/hip_runtime.h>
#include <hip/hip_bf16.h>
#include <math.h>

typedef __attribute__((ext_vector_type(2))) float v2f;
typedef __attribute__((ext_vector_type(8))) float v8f;

#define S_TOK 262144
#define DIM   1024
#define NEXP  64
#define TPB   256          // 8 waves (wave32)
#define TOK_PER_BLK 128    // 8 waves x 16 tokens
#define NBLK (S_TOK / TOK_PER_BLK)   // 2048
#define LGS  65            // LDS row stride (pad to dodge bank conflicts)

// ---------------------------------------------------------------------------
// Kernel 1: fp32 WMMA logits + per-token argmax/softmax + ordered in-block
// ranks + per-block expert histograms + per-block me[] partials.
// ---------------------------------------------------------------------------
__global__ __launch_bounds__(TPB)
void gate_logits_kernel(const float* __restrict__ x,
                        const float* __restrict__ W,
                        float* __restrict__ gates_out,   // [S]
                        int*   __restrict__ idx_out,     // [S]
                        int*   __restrict__ rank_ws,     // [S] in-block rank
                        float* __restrict__ mePart,      // [NBLK*NEXP]
                        int*   __restrict__ blkCnt)      // [NBLK*NEXP]
{
    __shared__ float lg[TOK_PER_BLK * LGS];
    __shared__ float maxv[TOK_PER_BLK];
    __shared__ float invs[TOK_PER_BLK];
    __shared__ int   hist[4 * NEXP];    // per-wave expert histogram (phase-2 waves 0..3)
    __shared__ int   wbase[4 * NEXP];   // exclusive scan of hist over waves

    const int tid  = threadIdx.x;
    const int wave = tid >> 5;
    const int lane = tid & 31;
    const int hi   = lane >> 4;        // 0: lanes 0-15, 1: lanes 16-31
    const int l15  = lane & 15;
    const int blk  = blockIdx.x;
    const long tok0 = (long)blk * TOK_PER_BLK + (long)wave * 16;

    // ---- Phase 1: WMMA fp32 16x16x4, one 16-token x 64-expert tile per wave.
    // A (16x4 f32) lane layout: lanes 0-15 hold K=0,1; lanes 16-31 hold K=2,3.
    // B (4x16 f32) lane layout mirrors it with N = lane&15 (B[k][n] = W[n][k]).
    v8f acc0 = {}, acc1 = {}, acc2 = {}, acc3 = {};
    const float* xrow = x + (tok0 + l15) * DIM + 2 * hi;
    const float* wrow = W + (long)l15 * DIM + 2 * hi;   // N-tiles at +16*DIM strides

    for (int k0 = 0; k0 < DIM; k0 += 32) {
        // one near-cache prefetch per 32 k-steps, 512B/lane ahead of consumption;
        // tail over-prefetch is speculative and silently dropped by hardware.
        __builtin_prefetch(xrow + k0 + 128, 0, 3);
        #pragma unroll
        for (int kk = 0; kk < 32; kk += 4) {
            const int k = k0 + kk;
            v2f a  = *(const v2f*)(xrow + k);
            v2f b0 = *(const v2f*)(wrow + k);
            v2f b1 = *(const v2f*)(wrow + 16 * DIM + k);
            v2f b2 = *(const v2f*)(wrow + 32 * DIM + k);
            v2f b3 = *(const v2f*)(wrow + 48 * DIM + k);
            acc0 = __builtin_amdgcn_wmma_f32_16x16x4_f32(false, a, false, b0, (short)0, acc0, false, false);
            acc1 = __builtin_amdgcn_wmma_f32_16x16x4_f32(false, a, false, b1, (short)0, acc1, false, false);
            acc2 = __builtin_amdgcn_wmma_f32_16x16x4_f32(false, a, false, b2, (short)0, acc2, false, false);
            acc3 = __builtin_amdgcn_wmma_f32_16x16x4_f32(false, a, false, b3, (short)0, acc3, false, false);
        }
    }

    // C/D layout: VGPR r, lanes 0-15 -> (M=r,   N=lane),
    //                      lanes16-31 -> (M=r+8, N=lane-16)
    #pragma unroll
    for (int r = 0; r < 8; ++r) {
        const int trow = wave * 16 + r + 8 * hi;
        lg[trow * LGS +  0 + l15] = acc0[r];
        lg[trow * LGS + 16 + l15] = acc1[r];
        lg[trow * LGS + 32 + l15] = acc2[r];
        lg[trow * LGS + 48 + l15] = acc3[r];
    }
    __syncthreads();

    // ---- Phase 2: per-token argmax + softmax denom (threads 0..127 = tokens)
    int myexp = 0;
    if (tid < TOK_PER_BLK) {
        const float* row = &lg[tid * LGS];
        float m = row[0];
        int   am = 0;
        #pragma unroll 8
        for (int e = 1; e < NEXP; ++e) {
            float v = row[e];
            if (v > m) { m = v; am = e; }   // strict '>' keeps first max (jnp.argmax)
        }
        float s = 0.f;
        #pragma unroll 8
        for (int e = 0; e < NEXP; ++e) s += __expf(row[e] - m);
        const float inv = 1.0f / s;         // = softmax prob at the argmax
        maxv[tid] = m;
        invs[tid] = inv;
        const long tok = (long)blk * TOK_PER_BLK + tid;
        gates_out[tok] = inv;
        idx_out[tok]   = am;
        myexp = am;
    }
    hist[tid] = 0;                          // 4*64 == 256 entries, one per thread
    __syncthreads();

    // ---- Phase 3: ordered rank within wave (wave32 shuffle scan)
    int rank = 0;
    if (tid < TOK_PER_BLK) {
        int after = 0;
        #pragma unroll
        for (int i = 0; i < 32; ++i) {
            int oe = __shfl(myexp, i, 32);
            if (oe == myexp) {
                if (i < lane) ++rank;
                else if (i > lane) ++after;
            }
        }
        if (after == 0) hist[wave * NEXP + myexp] = rank + 1;  // unique writer
    }
    __syncthreads();

    // ---- Phase 4: exclusive scan over the 4 waves, per-block totals, me[]
    if (tid < NEXP) {
        int run = 0;
        #pragma unroll
        for (int w = 0; w < 4; ++w) {
            wbase[w * NEXP + tid] = run;
            run += hist[w * NEXP + tid];
        }
        blkCnt[(long)blk * NEXP + tid] = run;   // ce partial for this block

        // deterministic per-expert softmax-mass partial (fixed summation order)
        float acc = 0.f;
        for (int t = 0; t < TOK_PER_BLK; ++t)
            acc += __expf(lg[t * LGS + tid] - maxv[t]) * invs[t];
        mePart[(long)blk * NEXP + tid] = acc;
    }
    __syncthreads();

    // ---- Phase 5: in-block slot index
    if (tid < TOK_PER_BLK) {
        const long tok = (long)blk * TOK_PER_BLK + tid;
        rank_ws[tok] = wbase[wave * NEXP + myexp] + rank;
    }
}

// ---------------------------------------------------------------------------
// Kernel 2: serial (deterministic) scan over blocks: me[e], ce[e], exclusive
// block offsets (in place), and l_aux.
// ---------------------------------------------------------------------------
__global__ __launch_bounds__(NEXP)
void gate_reduce_kernel(const float* __restrict__ mePart,
                        int* __restrict__ blkCnt,        // in: counts, out: excl offsets
                        float* __restrict__ l_aux_out)
{
    __shared__ float red[NEXP];
    const int e = threadIdx.x;
    float me = 0.f;
    int   run = 0;
    for (int b = 0; b < NBLK; ++b) {
        me += mePart[(long)b * NEXP + e];
        const int c = blkCnt[(long)b * NEXP + e];
        blkCnt[(long)b * NEXP + e] = run;    // exclusive prefix for kernel 3
        run += c;
    }
    red[e] = me * (float)run;                // me[e] * ce[e]
    __syncthreads();
    if (e == 0) {
        float s = 0.f;
        for (int i = 0; i < NEXP; ++i) s += red[i];
        l_aux_out[0] = s * ((float)NEXP / ((float)S_TOK * (float)S_TOK));
    }
}

// ---------------------------------------------------------------------------
// Kernel 3: final location = block offset + in-block rank.
// ---------------------------------------------------------------------------
__global__ __launch_bounds__(256)
void gate_scatter_kernel(const int* __restrict__ idx_out,
                         const int* __restrict__ rank_ws,
                         const int* __restrict__ blkOff,
                         int* __restrict__ loc_out)
{
    const long s = (long)blockIdx.x * 256 + threadIdx.x;
    const int  e = idx_out[s];
    const int  b = (int)(s / TOK_PER_BLK);
    loc_out[s] = blkOff[(long)b * NEXP + e] + rank_ws[s];
}

// ---------------------------------------------------------------------------
extern "C" void kernel_launch(void* const* d_in, const int* in_sizes, int n_in,
                              void* d_out, int out_size, void* d_ws, size_t ws_size,
                              hipStream_t stream) {
    const float* x = (const float*)d_in[0];   // [S, D]
    const float* W = (const float*)d_in[1];   // [E, D]

    // d_out flat: [ l_aux(1) | gates1_s(S) f32 | indices(S) i32 | locations(S) i32 ]
    float* o        = (float*)d_out;
    float* gates    = o + 1;
    int*   idx_out  = (int*)(o + 1 + S_TOK);
    int*   loc_out  = (int*)(o + 1 + 2 * (size_t)S_TOK);

    // workspace partition
    char*  ws      = (char*)d_ws;
    int*   rank_ws = (int*)ws;                                    // S ints
    float* mePart  = (float*)(ws + (size_t)S_TOK * 4);            // NBLK*NEXP f32
    int*   blkCnt  = (int*)(ws + (size_t)S_TOK * 4
                               + (size_t)NBLK * NEXP * 4);        // NBLK*NEXP i32

    gate_logits_kernel<<<NBLK, TPB, 0, stream>>>(x, W, gates, idx_out,
                                                 rank_ws, mePart, blkCnt);
    gate_reduce_kernel<<<1, NEXP, 0, stream>>>(mePart, blkCnt, o);
    gate_scatter_kernel<<<S_TOK / 256, 256, 0, stream>>>(idx_out, rank_ws,
                                                         blkCnt, loc_out);
}